// KWinnersTakeAll_39067022524560
// MI455X (gfx1250) — compile-verified
//
#include <hip/hip_runtime.h>
#include <math.h>

// ---------------------------------------------------------------------------
// k-Winners-Take-All forward mask for MI455X (gfx1250, wave32).
//
//   out[r, c] = (x[r,c] * boost[c]) >= kth_largest_row_r(x[r,:] * boost[:])
//   boost[c]  = expf(1.5f * (0.02f - duty[c]))
//   rows = 4096, cols = 16384, k = max(1, round(16384 * 0.02)) = 328
//
// Memory-bound: 512 MB streamed @ 23.3 TB/s ~= 22 us floor.
// One workgroup per row; row keys stay in VGPRs; exact k-th value found by a
// fully-unrolled 4-pass radix select over wave-privatized LDS histograms
// (ds_add_u32). x / out use NT cache hints so the 64 KB boost vector stays
// hot in the 192 MB L2 across all 4096 row-blocks. boost is precomputed once
// into d_ws (removes ~400 redundant VALU/thread of expf per row-block).
// ---------------------------------------------------------------------------

#define N_COLS    16384
#define K_WIN     328          // max(1, round(16384 * 0.02))
#define THREADS   512          // 16 wave32 waves
#define NWAVES    (THREADS / 32)
#define WH_STRIDE 257          // 257 % 64 == 1 -> wave-private hists hit distinct banks
#define VEC       8            // 8 x float4 = 32 elements per thread

typedef float v4f __attribute__((ext_vector_type(4)));

// Monotonic float -> uint key: unsigned order of keys == float order.
__device__ __forceinline__ unsigned int f2key(float f) {
  unsigned int u = __float_as_uint(f);
  return (u & 0x80000000u) ? ~u : (u | 0x80000000u);
}

// One-shot: boost[i] = expf(1.5f * (0.02f - duty[i]))  (64 KB, L2-resident)
__global__ __launch_bounds__(256)
void boost_precompute(const float* __restrict__ duty,
                      float* __restrict__ boost, int n) {
  const int i = blockIdx.x * blockDim.x + threadIdx.x;
  if (i < n) boost[i] = expf(1.5f * (0.02f - duty[i]));
}

// PRECOMP=true : second operand is the precomputed boost vector.
// PRECOMP=false: second operand is raw duty; expf computed inline (fallback).
template <bool PRECOMP>
__global__ __launch_bounds__(THREADS)
void kwta_topk_mask(const float* __restrict__ x,
                    const float* __restrict__ dob,   // duty or boost
                    float* __restrict__ out,
                    int rows) {
  __shared__ unsigned int whist[NWAVES * WH_STRIDE]; // wave-private histograms
  __shared__ unsigned int hist[256];                 // reduced histogram / scan ping
  __shared__ unsigned int hist2[256];                // scan pong
  __shared__ unsigned int s_prefix;                  // selected high bits of key
  __shared__ unsigned int s_kth;                     // remaining rank from the top

  const int tid  = threadIdx.x;
  const int wave = tid >> 5;
  const int row  = blockIdx.x;
  if (row >= rows) return;  // block-uniform; grid sized to rows exactly

  const size_t base = (size_t)row * (size_t)N_COLS;
  const float* __restrict__ xrow = x + base;
  float* __restrict__ orow = out + base;

  // ---- Load row once (NT b128), apply boost, keep order keys in VGPRs ----
  unsigned int key[VEC * 4];
#pragma unroll
  for (int i = 0; i < VEC; ++i) {
    const int col = i * (THREADS * 4) + (tid << 2);
    const v4f xv = __builtin_nontemporal_load(
        reinterpret_cast<const v4f*>(xrow + col));
    v4f bv;
    if (PRECOMP) {
      bv = *reinterpret_cast<const v4f*>(dob + col);       // L2-resident boost
    } else {
      const v4f dv = *reinterpret_cast<const v4f*>(dob + col);
      bv.x = expf(1.5f * (0.02f - dv.x));
      bv.y = expf(1.5f * (0.02f - dv.y));
      bv.z = expf(1.5f * (0.02f - dv.z));
      bv.w = expf(1.5f * (0.02f - dv.w));
    }
    key[i * 4 + 0] = f2key(xv.x * bv.x);
    key[i * 4 + 1] = f2key(xv.y * bv.y);
    key[i * 4 + 2] = f2key(xv.z * bv.z);
    key[i * 4 + 3] = f2key(xv.w * bv.w);
  }

  if (tid == 0) { s_prefix = 0u; s_kth = K_WIN; }

  // ---- 4-pass radix select, fully unrolled (pass-0 prefix test folds away) ----
#pragma unroll
  for (int pass = 0; pass < 4; ++pass) {
    const int shift = 24 - 8 * pass;
    const unsigned int pmask =
        (pass == 0) ? 0u : (0xFFFFFFFFu << (shift + 8));

    for (int j = tid; j < NWAVES * WH_STRIDE; j += THREADS) whist[j] = 0u;
    __syncthreads();  // clear done; s_prefix/s_kth from previous phase visible

    const unsigned int pref = s_prefix;
    const unsigned int kth  = s_kth;

    unsigned int* const mh = whist + wave * WH_STRIDE;
#pragma unroll
    for (int i = 0; i < VEC * 4; ++i) {
      const unsigned int k = key[i];
      if (pass == 0 || (k & pmask) == pref) {
        atomicAdd(&mh[(k >> shift) & 255u], 1u);  // ds_add_u32
      }
    }
    __syncthreads();

    // Reduce wave-private histograms into hist[256]
    if (tid < 256) {
      unsigned int s = 0;
#pragma unroll
      for (int w = 0; w < NWAVES; ++w) s += whist[w * WH_STRIDE + tid];
      hist[tid] = s;
    }
    __syncthreads();

    // Inclusive suffix sum (ping-pong, 1 barrier/step):
    //   hist[b] = #candidates with current byte >= b
    {
      unsigned int* src = hist;
      unsigned int* dst = hist2;
#pragma unroll 1
      for (int off = 1; off < 256; off <<= 1) {
        if (tid < 256) {
          unsigned int v = src[tid];
          if (tid + off < 256) v += src[tid + off];
          dst[tid] = v;
        }
        __syncthreads();
        unsigned int* t = src; src = dst; dst = t;
      }
      // 8 steps (even) -> final result is back in hist[]
    }

    // Unique bin b with cum[b] >= kth > cum[b+1] holds the k-th largest
    if (tid < 256) {
      const unsigned int c  = hist[tid];
      const unsigned int cn = (tid == 255) ? 0u : hist[tid + 1];
      if (c >= kth && cn < kth) {
        s_prefix = pref | ((unsigned int)tid << shift);
        s_kth    = kth - cn;
      }
    }
    __syncthreads();
  }

  // s_prefix is now the exact key of the k-th largest element of this row.
  const unsigned int thr = s_prefix;

  // ---- Emit 0/1 mask; >= keeps ties, matching the reference semantics ----
#pragma unroll
  for (int i = 0; i < VEC; ++i) {
    const int col = i * (THREADS * 4) + (tid << 2);
    v4f o;
    o.x = (key[i * 4 + 0] >= thr) ? 1.0f : 0.0f;
    o.y = (key[i * 4 + 1] >= thr) ? 1.0f : 0.0f;
    o.z = (key[i * 4 + 2] >= thr) ? 1.0f : 0.0f;
    o.w = (key[i * 4 + 3] >= thr) ? 1.0f : 0.0f;
    __builtin_nontemporal_store(o, reinterpret_cast<v4f*>(orow + col));
  }
}

extern "C" void kernel_launch(void* const* d_in, const int* in_sizes, int n_in,
                              void* d_out, int out_size, void* d_ws, size_t ws_size,
                              hipStream_t stream) {
  (void)n_in; (void)out_size;
  const float* x    = (const float*)d_in[0];   // (rows, 16384) f32
  const float* duty = (const float*)d_in[1];   // (16384,) f32
  float* out        = (float*)d_out;           // (rows, 16384) f32 mask

  const int rows = in_sizes[0] / N_COLS;       // 4096 for the reference shapes

  if (ws_size >= (size_t)N_COLS * sizeof(float)) {
    // Precompute boost once (tiny), then stream rows against it.
    float* boost = (float*)d_ws;
    boost_precompute<<<dim3((N_COLS + 255) / 256), dim3(256), 0, stream>>>(
        duty, boost, N_COLS);
    kwta_topk_mask<true><<<dim3(rows), dim3(THREADS), 0, stream>>>(
        x, boost, out, rows);
  } else {
    // Fallback: compute boost inline (identical math, more VALU).
    kwta_topk_mask<false><<<dim3(rows), dim3(THREADS), 0, stream>>>(
        x, duty, out, rows);
  }
}